// Signature_61993557950973
// MI455X (gfx1250) — compile-verified
//
#include <hip/hip_runtime.h>
#include <hip/hip_bf16.h>

// CDNA5 / gfx1250 path-signature kernel.
// Levels (offsets in the 4680-float signature): L1@0 (8), L2@8 (64), L3@72 (512), L4@584 (4096).

typedef __attribute__((ext_vector_type(2))) float v2f;
typedef __attribute__((ext_vector_type(8))) float v8f;

#define SIG_LEN 4680
#define O2 8
#define O3 72
#define O4 584

// ---------------------------------------------------------------------------
// Stage 1: per-(batch, chunk) sequential exp-Chen scan, state in LDS.
// dst[(n*m + c)*SIG_LEN + i]
// ---------------------------------------------------------------------------
__global__ __launch_bounds__(256)
void sig_chunk_kernel(const float* __restrict__ path, float* __restrict__ dst,
                      int m, int clen) {
    const int c = blockIdx.x, n = blockIdx.y, tid = threadIdx.x;
    __shared__ float S[SIG_LEN];
    __shared__ float zsh[8];

    for (int i = tid; i < SIG_LEN; i += 256) S[i] = 0.0f;

    const int start = c * clen;
    int steps = 511 - start;
    if (steps > clen) steps = clen;
    const float* pb = path + ((size_t)n * 512 + start) * 8;

    // Per-thread constant sub-indices for its 16 level-4 entries (idx = tid*16+e)
    const int i1 = tid >> 5;          // idx>>9
    const int i2 = (tid >> 2) & 7;    // (idx>>6)&7
    const int s2i = tid >> 2;         // idx>>6  (0..63)
    const int j3a = 2 * tid;          // level-3 entries handled by this thread
    __syncthreads();

    for (int t = 0; t < steps; ++t) {
        if (tid < 8) zsh[tid] = pb[(t + 1) * 8 + tid] - pb[t * 8 + tid];
        __syncthreads();

        // ---- phase 1: read old state, compute new values into registers ----
        const float zi1 = zsh[i1];
        const float zi2 = zsh[i2];
        float zq[8];
#pragma unroll
        for (int q = 0; q < 8; ++q) zq[q] = zsh[q];
        const float i3v0 = zsh[j3a & 7];
        const float i3v1 = zsh[(j3a + 1) & 7];
        const float S1v = S[i1];
        const float S2v = S[O2 + s2i];
        const float S3v0 = S[O3 + j3a];
        const float S3v1 = S[O3 + j3a + 1];

        float t4[16];
        const int base4 = tid * 16;
#pragma unroll
        for (int e = 0; e < 16; ++e) {
            const float zi3 = (e < 8) ? i3v0 : i3v1;
            const float S3v = (e < 8) ? S3v0 : S3v1;
            const float w  = zq[e & 7];          // z[i4]
            const float p2 = zi3 * w * 0.5f;     // z^2/2  at (i3,i4)
            const float p3 = zi2 * p2 * (1.0f / 3.0f); // z^3/6
            const float p4 = zi1 * p3 * 0.25f;   // z^4/24
            t4[e] = S[O4 + base4 + e] + S3v * w + S2v * p2 + S1v * p3 + p4;
        }
        // level-3 (2 entries per thread) -- all operands already in registers
        const float t30 = S3v0 + S2v * i3v0 + S1v * (zi2 * i3v0 * 0.5f)
                        + zi1 * zi2 * i3v0 * (1.0f / 6.0f);
        const float t31 = S3v1 + S2v * i3v1 + S1v * (zi2 * i3v1 * 0.5f)
                        + zi1 * zi2 * i3v1 * (1.0f / 6.0f);
        float t2v = 0.0f, t1v = 0.0f;
        if (tid < 64) t2v = S[O2 + tid] + S[tid >> 3] * zsh[tid & 7]
                          + zsh[tid >> 3] * zsh[tid & 7] * 0.5f;
        if (tid < 8)  t1v = S[tid] + zsh[tid];
        __syncthreads();

        // ---- phase 2: write back ----
#pragma unroll
        for (int e = 0; e < 16; ++e) S[O4 + base4 + e] = t4[e];
        S[O3 + j3a] = t30;
        S[O3 + j3a + 1] = t31;
        if (tid < 64) S[O2 + tid] = t2v;
        if (tid < 8)  S[tid] = t1v;
        __syncthreads();
    }

    float* o = dst + ((size_t)n * m + c) * SIG_LEN;
    for (int i = tid; i < SIG_LEN; i += 256) o[i] = S[i];
}

// ---------------------------------------------------------------------------
// Stage 2: Chen product of adjacent chunk signatures.
// Level-4 term is a 64x20x64 f32 GEMM done with V_WMMA_F32_16X16X4_F32.
// ---------------------------------------------------------------------------
__global__ __launch_bounds__(256)
void chen_combine_kernel(const float* __restrict__ src, float* __restrict__ dst,
                         int mIn, int mOut) {
    const int p = blockIdx.x, n = blockIdx.y, tid = threadIdx.x;
    __shared__ float Sh[SIG_LEN];
    __shared__ float Th[SIG_LEN];
    __shared__ float Am[64 * 20];   // [S3mat | kron(S1,I8) | S2vec | 0-pad]
    __shared__ float Bm[20 * 64];   // [kron(I8,T1); T3mat; T2vec; 0-pad]

    const float* sp = src + ((size_t)n * mIn + 2 * p) * SIG_LEN;
    const float* tp = sp + SIG_LEN;
    for (int i = tid; i < SIG_LEN; i += 256) { Sh[i] = sp[i]; Th[i] = tp[i]; }
    __syncthreads();

    for (int i = tid; i < 64 * 20; i += 256) {
        const int r = i / 20, k = i % 20;
        float v;
        if (k < 8)        v = Sh[O3 + r * 8 + k];                      // S3mat[r,k]
        else if (k < 16)  v = ((r & 7) == (k - 8)) ? Sh[r >> 3] : 0.f; // kron(S1,I8)
        else if (k == 16) v = Sh[O2 + r];                              // S2vec
        else              v = 0.f;
        Am[i] = v;
    }
    for (int i = tid; i < 20 * 64; i += 256) {
        const int k = i / 64, cc = i % 64;
        float v;
        if (k < 8)        v = ((cc >> 3) == k) ? Th[cc & 7] : 0.f;     // kron(I8,T1)
        else if (k < 16)  v = Th[O3 + (k - 8) * 64 + cc];              // T3mat
        else if (k == 16) v = Th[O2 + cc];                             // T2vec
        else              v = 0.f;
        Bm[i] = v;
    }
    __syncthreads();

    float* o = dst + ((size_t)n * mOut + p) * SIG_LEN;

    // levels 1..3: cheap scalar
    if (tid < 8)  o[tid] = Sh[tid] + Th[tid];
    if (tid < 64) o[O2 + tid] = Sh[O2 + tid] + Th[O2 + tid]
                              + Sh[tid >> 3] * Th[tid & 7];
#pragma unroll
    for (int e = 0; e < 2; ++e) {
        const int j = 2 * tid + e;
        o[O3 + j] = Sh[O3 + j] + Th[O3 + j]
                  + Sh[j >> 6] * Th[O2 + (j & 63)]
                  + Sh[O2 + (j >> 3)] * Th[j & 7];
    }

    // level 4: U4 = (S4 + T4) + A(64x20) x B(20x64), 16 tiles over 8 waves
    const int wave = tid >> 5, lane = tid & 31;
    const int half = lane >> 4, nn = lane & 15;
#pragma unroll
    for (int tt = 0; tt < 2; ++tt) {
        const int tile = wave * 2 + tt;
        const int tm = tile >> 2, tn = tile & 3;
        v8f c;
#pragma unroll
        for (int j = 0; j < 8; ++j) {
            const int M = tm * 16 + j + 8 * half;
            const int Nc = tn * 16 + nn;
            c[j] = Sh[O4 + M * 64 + Nc] + Th[O4 + M * 64 + Nc];
        }
#pragma unroll
        for (int k0 = 0; k0 < 20; k0 += 4) {
            const int ka = k0 + 2 * half;
            v2f a, b;
            a.x = Am[(tm * 16 + nn) * 20 + ka];      // A[m, ka]
            a.y = Am[(tm * 16 + nn) * 20 + ka + 1];  // A[m, ka+1]
            b.x = Bm[ka * 64 + tn * 16 + nn];        // B[ka, n]
            b.y = Bm[(ka + 1) * 64 + tn * 16 + nn];  // B[ka+1, n]
            c = __builtin_amdgcn_wmma_f32_16x16x4_f32(
                    false, a, false, b, (short)0, c, false, false);
        }
#pragma unroll
        for (int j = 0; j < 8; ++j) {
            const int M = tm * 16 + j + 8 * half;
            const int Nc = tn * 16 + nn;
            o[O4 + M * 64 + Nc] = c[j];
        }
    }
}

// ---------------------------------------------------------------------------
extern "C" void kernel_launch(void* const* d_in, const int* in_sizes, int n_in,
                              void* d_out, int out_size, void* d_ws, size_t ws_size,
                              hipStream_t stream) {
    const float* path = (const float*)d_in[0];
    float* out = (float*)d_out;
    const int N = 64;

    // choose number of chunks (power of two) that fits the workspace
    int m = 16;
    while (m > 1) {
        const size_t need = (size_t)(m + m / 2) * N * SIG_LEN * sizeof(float);
        if (need <= ws_size) break;
        m >>= 1;
    }

    if (m == 1) {
        // fully sequential fallback: one workgroup per batch, write d_out directly
        sig_chunk_kernel<<<dim3(1, N), 256, 0, stream>>>(path, out, 1, 511);
        return;
    }

    float* buf0 = (float*)d_ws;
    float* buf1 = buf0 + (size_t)m * N * SIG_LEN;
    const int clen = (511 + m - 1) / m;

    sig_chunk_kernel<<<dim3(m, N), 256, 0, stream>>>(path, buf0, m, clen);

    float* src = buf0;
    float* other = buf1;
    int cur = m;
    while (cur > 1) {
        const int outm = cur / 2;
        float* d = (outm == 1) ? out : other;
        chen_combine_kernel<<<dim3(outm, N), 256, 0, stream>>>(src, d, cur, outm);
        float* tmp = src; src = other; other = tmp;
        cur = outm;
    }
}